// GraphNet_56057913147714
// MI455X (gfx1250) — compile-verified
//
#include <hip/hip_runtime.h>

// ---------------------------------------------------------------------------
// GCN (edge-conditioned) on MI455X / gfx1250, f32 end-to-end.
// Matrix work uses V_WMMA_F32_16X16X4_F32 (exact f32, wave32).
// Edge kernel: WMMA edge-embedding + shuffle-broadcast metadata +
// coalesced gather + global_atomic_add_f32 scatter.
// ---------------------------------------------------------------------------

typedef __attribute__((ext_vector_type(2))) float v2f;
typedef __attribute__((ext_vector_type(4))) float v4f;
typedef __attribute__((ext_vector_type(8))) float v8f;

#define N_NODES  100000
#define N_EDGES  3200000
#define N_GRAPHS 64
#define IN_CH    128
#define HID      16
#define OUT_CH   8

__device__ __forceinline__ void gatomAdd(float* p, float v) {
  // lowers to global_atomic_add_f32 (no return)
  unsafeAtomicAdd(p, v);
}

// ---- degree / norm ---------------------------------------------------------
__global__ void k_init_deg(float* __restrict__ deg) {
  int i = blockIdx.x * blockDim.x + threadIdx.x;
  if (i < N_NODES) deg[i] = 1.0f;                 // self-loop contribution
}

__global__ void k_count_deg(const int* __restrict__ dst, float* __restrict__ deg) {
  int e = blockIdx.x * blockDim.x + threadIdx.x;
  if (e < N_EDGES) gatomAdd(&deg[dst[e]], 1.0f);
}

__global__ void k_dinv(float* __restrict__ deg) {
  int i = blockIdx.x * blockDim.x + threadIdx.x;
  if (i < N_NODES) {
    float d = deg[i];
    deg[i] = (d > 0.0f) ? rsqrtf(fmaxf(d, 1.0f)) : 0.0f;
  }
}

// ---- vectorized zero (global_store_b128) -----------------------------------
__global__ void k_zero4(v4f* __restrict__ p, int n4) {
  int i = blockIdx.x * blockDim.x + threadIdx.x;
  if (i < n4) p[i] = (v4f){0.f, 0.f, 0.f, 0.f};
}

// ---- node GEMM: Y[N,16] = X[N,K] @ W[K,16] via wmma f32 16x16x4 ------------
// One wave per 16-row tile. A layout: lane L -> row L&15, K = 2*(L>>4)+{0,1}.
// B layout: b.x = W[K=base + 2*(L>>4)][n=L&15], b.y = next K row.
// C layout: c[i] -> row (i + 8*(L>>4)), col L&15.
template <int K>
__global__ void __launch_bounds__(64) k_gemm_wmma(const float* __restrict__ X,
                                                  const float* __restrict__ W,
                                                  float* __restrict__ Y) {
  int wave = (int)((blockIdx.x * blockDim.x + threadIdx.x) >> 5); // tile id
  int lane = threadIdx.x & 31;
  int hi = lane >> 4;
  int m  = lane & 15;
  const float* xrow = X + (long)(wave * 16 + m) * K;
  v8f c = {};
#pragma unroll
  for (int kk = 0; kk < K; kk += 4) {
    int ka = kk + hi * 2;
    v2f a = *(const v2f*)(xrow + ka);
    v2f b;
    b.x = W[ka * 16 + m];
    b.y = W[(ka + 1) * 16 + m];
    c = __builtin_amdgcn_wmma_f32_16x16x4_f32(false, a, false, b,
                                              (short)0, c, false, false);
  }
  long base = (long)wave * 16;
#pragma unroll
  for (int i = 0; i < 8; ++i)
    Y[(base + i + hi * 8) * 16 + m] = c[i];
}

// ---- fused edge kernel -----------------------------------------------------
// One wave per 16-edge tile: C tile = EA[16x16] @ WE[16x16]; lane m owns the
// metadata of edge ebase+m (coalesced loads), broadcast per-edge via __shfl.
__global__ void __launch_bounds__(256) k_edge_msg(const float* __restrict__ EA,
                                                  const float* __restrict__ WE,
                                                  const int* __restrict__ SRC,
                                                  const int* __restrict__ DST,
                                                  const float* __restrict__ dinv,
                                                  const float* __restrict__ NODE,
                                                  float* __restrict__ ACC) {
  int wave = (int)((blockIdx.x * blockDim.x + threadIdx.x) >> 5);
  int lane = threadIdx.x & 31;
  int hi = lane >> 4;
  int m  = lane & 15;
  long ebase = (long)wave * 16;

  // per-lane metadata for edge (ebase + m); lanes L and L+16 hold duplicates
  int   sv   = SRC[ebase + m];
  int   dv   = DST[ebase + m];
  float nrmv = dinv[sv] * dinv[dv];

  // edge embedding tile via WMMA
  const float* row = EA + (ebase + m) * 16;
  v8f c = {};
#pragma unroll
  for (int kk = 0; kk < 16; kk += 4) {
    int ka = kk + hi * 2;
    v2f a = *(const v2f*)(row + ka);
    v2f b;
    b.x = WE[ka * 16 + m];
    b.y = WE[(ka + 1) * 16 + m];
    c = __builtin_amdgcn_wmma_f32_16x16x4_f32(false, a, false, b,
                                              (short)0, c, false, false);
  }
  // c[i] = edge_emb[edge = ebase + i + 8*hi][channel = m]
#pragma unroll
  for (int i = 0; i < 8; ++i) {
    int el = i + hi * 8;                       // which edge-in-tile this lane needs
    int   s   = __shfl(sv, el, 32);
    int   d   = __shfl(dv, el, 32);
    float nrm = __shfl(nrmv, el, 32);
    float msg = nrm * (NODE[(long)s * 16 + m] + c[i]); // coalesced 64B gather
    gatomAdd(&ACC[(long)d * 16 + m], msg);             // coalesced 16-ch scatter
  }
}

// ---- finalize: add analytic self-loop term and ReLU (float4) ---------------
__global__ void k_finalize_relu(const v4f* __restrict__ ACC,
                                const v4f* __restrict__ NODE,
                                const float* __restrict__ dinv,
                                v4f* __restrict__ H) {
  int i = blockIdx.x * blockDim.x + threadIdx.x;   // over N*4 float4 groups
  if (i < N_NODES * 4) {
    int v = i >> 2;                                // 4 groups per node
    float di = dinv[v];
    float d2 = di * di;                            // self-loop: norm = dinv^2
    v4f a = ACC[i];
    v4f n = NODE[i];
    v4f r;
#pragma unroll
    for (int k = 0; k < 4; ++k) {
      float val = a[k] + d2 * n[k];
      r[k] = val > 0.0f ? val : 0.0f;
    }
    H[i] = r;
  }
}

// ---- mean pool: LDS block reduction (ds_add_f32), then global atomics ------
__global__ void __launch_bounds__(256) k_pool(const float* __restrict__ H,
                                              const int* __restrict__ batch,
                                              float* __restrict__ pooled,
                                              float* __restrict__ counts) {
  __shared__ float sg[N_GRAPHS * HID];
  __shared__ float sc[N_GRAPHS];
  int tid = threadIdx.x;
  for (int i = tid; i < N_GRAPHS * HID; i += blockDim.x) sg[i] = 0.0f;
  for (int i = tid; i < N_GRAPHS; i += blockDim.x) sc[i] = 0.0f;
  __syncthreads();
  int v = blockIdx.x * blockDim.x + tid;
  if (v < N_NODES) {
    int g = batch[v];
    atomicAdd(&sc[g], 1.0f);
    const v4f* hrow = (const v4f*)(H + (long)v * 16);
#pragma unroll
    for (int q = 0; q < 4; ++q) {
      v4f hv = hrow[q];
#pragma unroll
      for (int k = 0; k < 4; ++k)
        atomicAdd(&sg[g * HID + q * 4 + k], hv[k]);
    }
  }
  __syncthreads();
  for (int i = tid; i < N_GRAPHS * HID; i += blockDim.x) gatomAdd(&pooled[i], sg[i]);
  for (int i = tid; i < N_GRAPHS; i += blockDim.x) gatomAdd(&counts[i], sc[i]);
}

// ---- head: out[64,8] = (pooled / counts) @ W[16,8] + b ---------------------
__global__ void k_head(const float* __restrict__ pooled,
                       const float* __restrict__ counts,
                       const float* __restrict__ W,
                       const float* __restrict__ b,
                       float* __restrict__ out) {
  int t = threadIdx.x;                              // 512 threads, exact
  if (t < N_GRAPHS * OUT_CH) {
    int g = t / OUT_CH, o = t % OUT_CH;
    float inv = 1.0f / fmaxf(counts[g], 1.0f);
    float acc = b[o];
#pragma unroll
    for (int c = 0; c < HID; ++c)
      acc += pooled[g * HID + c] * inv * W[c * OUT_CH + o];
    out[t] = acc;
  }
}

// ---------------------------------------------------------------------------
extern "C" void kernel_launch(void* const* d_in, const int* in_sizes, int n_in,
                              void* d_out, int out_size, void* d_ws, size_t ws_size,
                              hipStream_t stream) {
  const float* x     = (const float*)d_in[0];
  const int*   ei    = (const int*)  d_in[1];     // [2,E] row-major
  const float* ea    = (const float*)d_in[2];
  const int*   batch = (const int*)  d_in[3];
  const float* W1n   = (const float*)d_in[4];
  const float* W1e   = (const float*)d_in[5];
  const float* W2n   = (const float*)d_in[6];
  const float* W2e   = (const float*)d_in[7];
  const float* nW    = (const float*)d_in[8];
  const float* nb    = (const float*)d_in[9];
  float* out = (float*)d_out;

  // workspace layout (floats): ~19.6 MB total
  float* ws     = (float*)d_ws;
  float* dinv   = ws;                       // N       (deg -> dinv in place)
  float* bufA   = ws + 102400;              // N*16    node_emb (pre-aggregation)
  float* bufB   = bufA + 1600000;           // N*16    scatter accumulator
  float* bufC   = bufB + 1600000;           // N*16    h (post-ReLU)
  float* pooled = bufC + 1600000;           // 64*16
  float* counts = pooled + N_GRAPHS * HID;  // 64

  const int* SRC = ei;
  const int* DST = ei + N_EDGES;

  dim3 b256(256);
  const int gemmBlocks = N_NODES / 16 / 2;      // 3125 blocks x 2 waves
  const int edgeBlocks = N_EDGES / 16 / 8;      // 25000 blocks x 8 waves
  const int zero4Blk   = (N_NODES * 4 + 255) / 256;   // N*16 floats = N*4 float4
  const int finBlk     = (N_NODES * 4 + 255) / 256;

  // symmetric gcn_norm
  k_init_deg <<<(N_NODES + 255) / 256, b256, 0, stream>>>(dinv);
  k_count_deg<<<(N_EDGES + 255) / 256, b256, 0, stream>>>(DST, dinv);
  k_dinv     <<<(N_NODES + 255) / 256, b256, 0, stream>>>(dinv);

  // ---- layer 1 ----
  k_gemm_wmma<IN_CH><<<gemmBlocks, dim3(64), 0, stream>>>(x, W1n, bufA);
  k_zero4<<<zero4Blk, b256, 0, stream>>>((v4f*)bufB, N_NODES * 4);
  k_edge_msg<<<edgeBlocks, b256, 0, stream>>>(ea, W1e, SRC, DST, dinv, bufA, bufB);
  k_finalize_relu<<<finBlk, b256, 0, stream>>>((const v4f*)bufB, (const v4f*)bufA,
                                               dinv, (v4f*)bufC);

  // ---- layer 2 ----
  k_gemm_wmma<HID><<<gemmBlocks, dim3(64), 0, stream>>>(bufC, W2n, bufA);
  k_zero4<<<zero4Blk, b256, 0, stream>>>((v4f*)bufB, N_NODES * 4);
  k_edge_msg<<<edgeBlocks, b256, 0, stream>>>(ea, W2e, SRC, DST, dinv, bufA, bufB);
  k_finalize_relu<<<finBlk, b256, 0, stream>>>((const v4f*)bufB, (const v4f*)bufA,
                                               dinv, (v4f*)bufC);

  // ---- pool + head ----
  k_zero4<<<1, b256, 0, stream>>>((v4f*)pooled, (N_GRAPHS * HID + N_GRAPHS) / 4);
  k_pool<<<(N_NODES + 255) / 256, b256, 0, stream>>>(bufC, batch, pooled, counts);
  k_head<<<1, dim3(512), 0, stream>>>(pooled, counts, nW, nb, out);
}